// MultiheadAttentionLocal_37297495998528
// MI455X (gfx1250) — compile-verified
//
#include <hip/hip_runtime.h>
#include <hip/hip_bf16.h>
#include <math.h>

// Problem constants (match reference)
#define NQ 4096
#define MK 16384
#define NB 8
#define LN 96
#define CC 512
#define NH 8
#define DH 64   // CC / NH

typedef float v2f __attribute__((ext_vector_type(2)));
typedef float v4f __attribute__((ext_vector_type(4)));
typedef float v8f __attribute__((ext_vector_type(8)));
typedef uint32_t u32x4 __attribute__((ext_vector_type(4)));
typedef int32_t  i32x4 __attribute__((ext_vector_type(4)));
typedef int32_t  i32x8 __attribute__((ext_vector_type(8)));

#if __has_builtin(__builtin_amdgcn_tensor_load_to_lds)
#define HAVE_TDM 1
#else
#define HAVE_TDM 0
#endif

// ---------------------------------------------------------------------------
// CDNA5 async global->LDS copy (ASYNCcnt-tracked). VDST = LDS byte address
// (low 32 bits of a generic __shared__ pointer == workgroup-relative offset).
// ---------------------------------------------------------------------------
__device__ __forceinline__ void async_load_lds_b128(uint32_t lds_addr, const float* g)
{
    asm volatile("global_load_async_to_lds_b128 %0, %1, off"
                 :: "v"(lds_addr), "v"((unsigned long long)(uintptr_t)g)
                 : "memory");
}
__device__ __forceinline__ void wait_async_le2() {
    asm volatile("s_wait_asynccnt 0x2" ::: "memory");
}
__device__ __forceinline__ void wait_async_le4() {
    asm volatile("s_wait_asynccnt 0x4" ::: "memory");
}
__device__ __forceinline__ void wait_async_0() {
    asm volatile("s_wait_asynccnt 0x0" ::: "memory");
}

#define GT_M 64
#define GT_N 64
#define GT_K 16
#define NCHUNK (CC / GT_K)   // 32
#define LDSS 20              // padded row stride: conflict-free, 16B-aligned

#if HAVE_TDM
// ---------------------------------------------------------------------------
// TDM: DMA one 64-row x 16-float tile of W (row stride 512 floats) into LDS.
// D# per CDNA5 ISA:
//  group0: [1:0]=count=1 | [63:32]=lds_addr | [120:64]=global_addr | [127:126]=type 2
//  group1: data_size=2(4B), pad_enable, pad_interval=3 (16 DW), pad_amount=3 (4 DW)
//          -> LDS row stride = 20 floats (matches LDSS); tensor 512x512,
//          tile_dim0=16 (along rows), tile_dim1=64 (rows), dim0_stride=512.
//  groups 2/3: zero (2-D tensor). Tracked on TENSORcnt.
// 6-arg clang-23 builtin: (u32x4, i32x8, i32x4, i32x4, i32x8, i32 cpol).
// ---------------------------------------------------------------------------
__device__ __forceinline__ void tdm_load_w_tile(const float* g, uint32_t lds_addr)
{
    const unsigned long long ga = (unsigned long long)(uintptr_t)g;
    u32x4 g0 = { 1u,                                   // valid user descriptor
                 lds_addr,                             // LDS byte address
                 (uint32_t)ga,                         // global_addr[31:0]
                 (uint32_t)((ga >> 32) & 0x01FFFFFFu) | 0x80000000u }; // [56:32] | type=2
    i32x8 g1 = { (int)((2u << 16) | (1u << 20) | (3u << 22) | (3u << 25)),
                 (int)(512u << 16),    // tensor_dim0 = 512 (low16 -> bits 63:48)
                 (int)(512u << 16),    // dim0 hi = 0; tensor_dim1 low16 = 512
                 (int)(16u << 16),     // tensor_dim1 hi = 0; tile_dim0 = 16
                 64,                   // tile_dim1 = 64, tile_dim2 = 0
                 512,                  // tensor_dim0_stride = 512 (low 32)
                 0, 0 };
    i32x4 z4 = { 0, 0, 0, 0 };
    i32x8 z8 = { 0, 0, 0, 0, 0, 0, 0, 0 };
    __builtin_amdgcn_tensor_load_to_lds(g0, g1, z4, z4, z8, 0);
}
#endif

// ---------------------------------------------------------------------------
// Tiled fp32 WMMA GEMM:  Y[rows x 512] = (X[rows x 512] @ W[512 x 512]^T + b) * scale
// Block: 128 threads (4 waves). Block tile 64x64; each wave owns a 16(M)x64(N)
// strip = 4 accumulators of 16x16. K in 16-wide chunks, DOUBLE-BUFFERED in LDS:
//   A tile : per-thread global_load_async_to_lds_b128   (ASYNCcnt)
//   W tile : one TDM tensor_load_to_lds per block/chunk (TENSORcnt, wave 0)
// staging of chunk i+1 overlaps the 16 v_wmma_f32_16x16x4_f32 of chunk i.
// ---------------------------------------------------------------------------
__global__ __launch_bounds__(128)
void gemm512_wmma_f32(const float* __restrict__ X,
                      const float* __restrict__ W,
                      const float* __restrict__ bias,
                      float* __restrict__ Y,
                      float scale)
{
    __shared__ float As[2][GT_M * LDSS];
    __shared__ float Ws[2][GT_N * LDSS];

    const int t    = threadIdx.x;       // 0..127
    const int wave = t >> 5;            // 0..3 -> M strip
    const int lane = t & 31;
    const int idx  = lane & 15;         // M (A frag) / N (B frag) within tile
    const int half = lane >> 4;         // selects K pair within frag

    const int row0 = blockIdx.x * GT_M;
    const int col0 = blockIdx.y * GT_N;

    // A staging: each thread owns 8 floats (two b128) per chunk
    const int lr = t >> 1;              // 0..63 : row within tile
    const int lc = (t & 1) * 8;         // 0 or 8 : col within 16-wide K chunk

    const float* aB = X + (size_t)(row0 + lr) * CC + lc;   // + k0 per chunk
    const uint32_t ldsA0 = (uint32_t)(uintptr_t)&As[0][lr * LDSS + lc];
    const uint32_t ldsA1 = (uint32_t)(uintptr_t)&As[1][lr * LDSS + lc];
#if HAVE_TDM
    const float* wT = W + (size_t)col0 * CC;               // W tile origin (+k0 per chunk)
    const uint32_t ldsWt0 = (uint32_t)(uintptr_t)&Ws[0][0];
    const uint32_t ldsWt1 = (uint32_t)(uintptr_t)&Ws[1][0];
#else
    const float* wB = W + (size_t)(col0 + lr) * CC + lc;
    const uint32_t ldsW0 = (uint32_t)(uintptr_t)&Ws[0][lr * LDSS + lc];
    const uint32_t ldsW1 = (uint32_t)(uintptr_t)&Ws[1][lr * LDSS + lc];
#endif

    v8f acc[4] = {};                    // 4 x (16x16 f32) accumulators

    // prologue: chunk 0 -> buffer 0
    async_load_lds_b128(ldsA0,      aB);
    async_load_lds_b128(ldsA0 + 16, aB + 4);
#if HAVE_TDM
    if (wave == 0) tdm_load_w_tile(wT, ldsWt0);
#else
    async_load_lds_b128(ldsW0,      wB);
    async_load_lds_b128(ldsW0 + 16, wB + 4);
#endif

    for (int i = 0; i < NCHUNK; ++i) {
        const int cur = i & 1;
        if (i + 1 < NCHUNK) {
            // overlap: stage chunk i+1 into the other buffer
            const float* an = aB + (i + 1) * GT_K;
            const uint32_t la = cur ? ldsA0 : ldsA1;
            async_load_lds_b128(la,      an);
            async_load_lds_b128(la + 16, an + 4);
#if HAVE_TDM
            if (wave == 0) tdm_load_w_tile(wT + (i + 1) * GT_K, cur ? ldsWt0 : ldsWt1);
            wait_async_le2();   // chunk i's A landed (in-order), chunk i+1 in flight
            if (wave == 0) __builtin_amdgcn_s_wait_tensorcnt((short)1); // chunk i's W landed
#else
            const float* wn = wB + (i + 1) * GT_K;
            const uint32_t lw = cur ? ldsW0 : ldsW1;
            async_load_lds_b128(lw,      wn);
            async_load_lds_b128(lw + 16, wn + 4);
            wait_async_le4();
#endif
        } else {
            wait_async_0();
#if HAVE_TDM
            if (wave == 0) __builtin_amdgcn_s_wait_tensorcnt((short)0);
#endif
        }
        __syncthreads();        // publish chunk i's LDS deposits to all waves

        const float* __restrict__ Ab = As[cur];
        const float* __restrict__ Wb = Ws[cur];
        #pragma unroll
        for (int kk = 0; kk < 4; ++kk) {
            const int kf = kk * 4 + half * 2;
            v2f a = *(const v2f*)&Ab[(wave * 16 + idx) * LDSS + kf];
            #pragma unroll
            for (int c2 = 0; c2 < 4; ++c2) {
                v2f b = *(const v2f*)&Wb[(c2 * 16 + idx) * LDSS + kf];
                acc[c2] = __builtin_amdgcn_wmma_f32_16x16x4_f32(
                    false, a, false, b, (short)0, acc[c2], false, false);
            }
        }
        __syncthreads();        // all waves done reading buf[cur] before reuse
    }

    // Epilogue: C/D layout -> VGPR r: lanes 0-15 give M=r, lanes 16-31 give M=r+8
    #pragma unroll
    for (int c2 = 0; c2 < 4; ++c2) {
        const int col  = col0 + c2 * 16 + idx;
        const float bc = bias[col];
        #pragma unroll
        for (int r = 0; r < 8; ++r) {
            const int row = row0 + wave * 16 + half * 8 + r;
            Y[(size_t)row * CC + col] = (acc[c2][r] + bc) * scale;
        }
    }
}

// ---------------------------------------------------------------------------
// Local attention core. One block (256 thr / 8 waves) per query.
// K/V projections total 64MB -> resident in MI455X's 192MB L2, so the gathered
// streams are L2 hits. Scores: 768 dot-64. Softmax per head by one wave using
// 32-lane shuffles (wave32). Writes ctx (pre out-proj) and attn-mean output.
// ---------------------------------------------------------------------------
__global__ __launch_bounds__(256)
void local_attn_kernel(const float* __restrict__ qp,   // (N, C) pre-scaled
                       const float* __restrict__ kp,   // (M, C)
                       const float* __restrict__ vp,   // (M, C)
                       const int*   __restrict__ index_pair,      // (N, L)
                       const int*   __restrict__ key_batch_cnt,   // (B)
                       const int*   __restrict__ index_pair_batch,// (N)
                       float* __restrict__ ctx,        // (N, C)
                       float* __restrict__ attn_mean)  // (N, L)
{
    const int n = blockIdx.x;
    const int t = threadIdx.x;

    __shared__ float qs[CC];
    __shared__ int   gidx[LN];
    __shared__ float sc[LN * NH];
    __shared__ float ps[LN * NH];

    // stage q row
    qs[t]       = qp[(size_t)n * CC + t];
    qs[t + 256] = qp[(size_t)n * CC + t + 256];

    // local->global key index translation
    if (t < LN) {
        const int b = index_pair_batch[n];
        int off = 0;
        for (int i = 0; i < NB; ++i) off += (i < b) ? key_batch_cnt[i] : 0;
        const int ip = index_pair[(size_t)n * LN + t];
        gidx[t] = (ip < 0) ? -1 : (ip + off);
    }
    __syncthreads();

    // scores: pair p = l*8 + h
    for (int p = t; p < LN * NH; p += 256) {
        const int l = p >> 3;
        const int h = p & 7;
        const int g = gidx[l];
        float s = -INFINITY;
        if (g >= 0) {
            const float* kr = kp + (size_t)g * CC + h * DH;
            const float* qr = qs + h * DH;
            float a = 0.f;
            #pragma unroll
            for (int d = 0; d < DH; d += 4) {
                v4f kv = *(const v4f*)(kr + d);
                a += kv.x * qr[d] + kv.y * qr[d + 1] + kv.z * qr[d + 2] + kv.w * qr[d + 3];
            }
            s = a;
        }
        sc[l * NH + h] = s;
    }
    __syncthreads();

    // softmax over L per head: wave w handles head w; lane owns l, l+32, l+64
    {
        const int h  = t >> 5;
        const int ln = t & 31;
        float x0 = sc[(ln)      * NH + h];
        float x1 = sc[(ln + 32) * NH + h];
        float x2 = sc[(ln + 64) * NH + h];
        float mx = fmaxf(fmaxf(x0, x1), x2);
        #pragma unroll
        for (int o = 16; o > 0; o >>= 1) mx = fmaxf(mx, __shfl_xor(mx, o, 32));
        float e0 = __expf(x0 - mx);
        float e1 = __expf(x1 - mx);
        float e2 = __expf(x2 - mx);
        float sm = e0 + e1 + e2;
        #pragma unroll
        for (int o = 16; o > 0; o >>= 1) sm += __shfl_xor(sm, o, 32);
        const float inv = 1.0f / sm;
        ps[(ln)      * NH + h] = e0 * inv;
        ps[(ln + 32) * NH + h] = e1 * inv;
        ps[(ln + 64) * NH + h] = e2 * inv;
    }
    __syncthreads();

    // attention-probability mean over heads (second output)
    if (t < LN) {
        float s = 0.f;
        #pragma unroll
        for (int h = 0; h < NH; ++h) s += ps[t * NH + h];
        attn_mean[(size_t)n * LN + t] = s * (1.0f / NH);
    }

    // weighted V sum: thread owns channels t and t+256 (coalesced v rows)
    {
        float a0 = 0.f, a1 = 0.f;
        const int h0 = t >> 6;            // head of channel t
        const int h1 = (t + 256) >> 6;    // head of channel t+256
        for (int l = 0; l < LN; ++l) {
            const int g = gidx[l];
            if (g < 0) continue;
            const float* vr = vp + (size_t)g * CC;
            a0 += ps[l * NH + h0] * vr[t];
            a1 += ps[l * NH + h1] * vr[t + 256];
        }
        ctx[(size_t)n * CC + t]       = a0;
        ctx[(size_t)n * CC + t + 256] = a1;
    }
}

// ---------------------------------------------------------------------------
extern "C" void kernel_launch(void* const* d_in, const int* in_sizes, int n_in,
                              void* d_out, int out_size, void* d_ws, size_t ws_size,
                              hipStream_t stream)
{
    const float* query      = (const float*)d_in[0];
    const float* key        = (const float*)d_in[1];
    const float* value      = (const float*)d_in[2];
    const int*   index_pair = (const int*)d_in[3];   // harness delivers integer inputs as int32
    /* d_in[4] query_batch_cnt unused */
    const int*   key_cnt    = (const int*)d_in[5];
    const int*   ip_batch   = (const int*)d_in[6];
    const float* in_w       = (const float*)d_in[7]; // (3C, C) row-major
    const float* in_b       = (const float*)d_in[8];
    const float* out_w      = (const float*)d_in[9]; // (C, C)
    const float* out_b      = (const float*)d_in[10];

    float* out       = (float*)d_out;                 // (N, C)
    float* attn_mean = out + (size_t)NQ * CC;         // (N, L)

    // workspace: qp | kp | vp | ctx  (80 MB total)
    float* qp  = (float*)d_ws;
    float* kp  = qp + (size_t)NQ * CC;
    float* vp  = kp + (size_t)MK * CC;
    float* ctx = vp + (size_t)MK * CC;

    const float scaling = 0.125f;  // Dh^-0.5 = 1/8 (applied after bias, as in ref)

    dim3 blk(128);
    // Q / K / V projections (fp32 WMMA, async + TDM double-buffered staging)
    gemm512_wmma_f32<<<dim3(NQ / GT_M, CC / GT_N), blk, 0, stream>>>(
        query, in_w, in_b, qp, scaling);
    gemm512_wmma_f32<<<dim3(MK / GT_M, CC / GT_N), blk, 0, stream>>>(
        key, in_w + (size_t)CC * CC, in_b + CC, kp, 1.0f);
    gemm512_wmma_f32<<<dim3(MK / GT_M, CC / GT_N), blk, 0, stream>>>(
        value, in_w + (size_t)2 * CC * CC, in_b + 2 * CC, vp, 1.0f);

    // gathered attention (scores, softmax, weighted sum, attn-mean output)
    local_attn_kernel<<<dim3(NQ), dim3(256), 0, stream>>>(
        qp, kp, vp, index_pair, key_cnt, ip_batch, ctx, attn_mean);

    // output projection straight into d_out
    gemm512_wmma_f32<<<dim3(NQ / GT_M, CC / GT_N), blk, 0, stream>>>(
        ctx, out_w, out_b, out, 1.0f);
}